// FlowUpSampler_5557687681310
// MI455X (gfx1250) — compile-verified
//
#include <hip/hip_runtime.h>
#include <stdint.h>

// Reference constants: N=8, H=64, W=128, 8x upsample, 3x3 kernel.
#define H_     64
#define W_     128
#define UPF    8
#define NT     128            // threads per block (= W), 4 wave32s
#define DEPTH  4              // async pipeline slots
#define LPAD   12             // taps per lane in LDS, padded 9 -> 12 (48 B/lane)
#define STAGES 8              // b = 0..7 per block (fixed a)

typedef float v4f __attribute__((ext_vector_type(4)));

#if __has_builtin(__builtin_amdgcn_exp2f)
#define EXP2F(x) __builtin_amdgcn_exp2f(x)
#else
#define EXP2F(x) exp2f(x)
#endif
#if __has_builtin(__builtin_amdgcn_rcpf)
#define RCPF(x) __builtin_amdgcn_rcpf(x)
#else
#define RCPF(x) (1.0f / (x))
#endif

// Async global->LDS DWORD copy (ASYNCcnt-tracked); per-stage step folded into
// the 24-bit immediate offset so the steady state needs no address VALU.
template<int OFF>
__device__ __forceinline__ void async_ld_f32(uint32_t lds_addr, const float* g) {
    asm volatile("global_load_async_to_lds_b32 %0, %1, off offset:%2 th:TH_LOAD_NT"
                 :
                 : "v"(lds_addr), "v"(g), "i"(OFF)
                 : "memory");
}

template<int N>
__device__ __forceinline__ void wait_async() {
    if constexpr (N >= 36)      asm volatile("s_wait_asynccnt 36" ::: "memory");
    else if constexpr (N == 27) asm volatile("s_wait_asynccnt 27" ::: "memory");
    else if constexpr (N == 18) asm volatile("s_wait_asynccnt 18" ::: "memory");
    else if constexpr (N == 9)  asm volatile("s_wait_asynccnt 9"  ::: "memory");
    else                        asm volatile("s_wait_asynccnt 0"  ::: "memory");
}

__device__ __forceinline__ void wait_ds0() {
    asm volatile("s_wait_dscnt 0" ::: "memory");
}

// Issue the 9 mask taps of stage S (relative b = S) into LDS slot S % DEPTH.
// LDS layout per slot: [lane w][tap k] so each lane later reads contiguously.
template<int S>
__device__ __forceinline__ void issue_stage(uint32_t ldsl, const float* (&pk)[9]) {
    constexpr uint32_t slotOff = (uint32_t)(S & (DEPTH - 1)) * (uint32_t)(NT * LPAD * 4);
#pragma unroll
    for (int k = 0; k < 9; ++k)
        async_ld_f32<S * (H_ * W_ * 4)>(ldsl + slotOff + (uint32_t)k * 4u, pk[k]);
}

template<int I>
__device__ __forceinline__ void do_stage(const float* sbase, int w, uint32_t ldsl,
                                         const float* (&pk)[9],
                                         const float (&p0)[9], const float (&p1)[9],
                                         float& out0, float& out1) {
    // Stages issued so far: 0 .. min(STAGES-1, I+DEPTH-1). Wait until stage I done.
    constexpr int last = (I + DEPTH - 1 <= STAGES - 1) ? (I + DEPTH - 1) : (STAGES - 1);
    wait_async<(last - I) * 9>();

    // Pull this lane's 9 taps into registers: b128 + b128 + b32.
    const float* bp = sbase + (I & (DEPTH - 1)) * (NT * LPAD) + w * LPAD;
    const v4f mA = *(const v4f*)(bp);
    const v4f mB = *(const v4f*)(bp + 4);
    const float m8 = bp[8];
    // Ensure the DS reads completed before the async engine may overwrite slot I.
    wait_ds0();
    if constexpr (I + DEPTH < STAGES) issue_stage<I + DEPTH>(ldsl, pk);

    const float m[9] = {mA.x, mA.y, mA.z, mA.w, mB.x, mB.y, mB.z, mB.w, m8};
    float s = 0.0f, a0 = 0.0f, a1 = 0.0f;
#pragma unroll
    for (int k = 0; k < 9; ++k) {
        const float e = EXP2F(m[k] * 1.4426950408889634f);  // exp(m)
        s += e;
        a0 = fmaf(e, p0[k], a0);
        a1 = fmaf(e, p1[k], a1);
    }
    const float r = RCPF(s);
    out0 = a0 * r;
    out1 = a1 * r;
}

__global__ __launch_bounds__(NT) void FlowUpSampler_kernel(
    const float* __restrict__ flow,   // [N, 2, H, W]
    const float* __restrict__ mask,   // [N, 576, H, W]
    float* __restrict__ out)          // [N, 2, 8H, 8W]
{
    __shared__ float buf[DEPTH][NT][LPAD];   // 24 KB

    const int w = threadIdx.x;                 // 0..127
    const int a = blockIdx.x & (UPF - 1);      // sub-row 0..7
    const int h = (blockIdx.x >> 3) & (H_ - 1);
    const int n = blockIdx.x >> 9;
    const int HW = H_ * W_;

    // ---- 3x3 neighborhood of 8*flow, zero-padded, both channels ----
    const float* f0 = flow + (size_t)(n * 2) * HW;
    const float* f1 = f0 + HW;
    float p0[9], p1[9];
#pragma unroll
    for (int di = 0; di < 3; ++di) {
#pragma unroll
        for (int dj = 0; dj < 3; ++dj) {
            const int hh = h + di - 1;
            const int ww = w + dj - 1;
            const bool ok = (hh >= 0) && (hh < H_) && (ww >= 0) && (ww < W_);
            const int idx = ok ? (hh * W_ + ww) : 0;
            const float sc = ok ? 8.0f : 0.0f;
            p0[di * 3 + dj] = sc * f0[idx];
            p1[di * 3 + dj] = sc * f1[idx];
        }
    }

    // Per-tap base pointers at b=0: channel = k*64 + a*8; stage b steps by HW*4 B
    // which is applied as the asm immediate offset.
    const float* pk[9];
    const float* mb = mask + ((size_t)n * 576 + (size_t)a * UPF) * HW
                    + (size_t)h * W_ + w;
#pragma unroll
    for (int k = 0; k < 9; ++k) pk[k] = mb + (size_t)(k * 64) * HW;

    const uint32_t ldsl = (uint32_t)(uintptr_t)&buf[0][w][0];
    const float* sbase = &buf[0][0][0];

    // Prologue: fill all DEPTH slots (36 async loads in flight, cnt max 63).
    issue_stage<0>(ldsl, pk);
    issue_stage<1>(ldsl, pk);
    issue_stage<2>(ldsl, pk);
    issue_stage<3>(ldsl, pk);

    float o0[STAGES], o1[STAGES];
    do_stage<0>(sbase, w, ldsl, pk, p0, p1, o0[0], o1[0]);
    do_stage<1>(sbase, w, ldsl, pk, p0, p1, o0[1], o1[1]);
    do_stage<2>(sbase, w, ldsl, pk, p0, p1, o0[2], o1[2]);
    do_stage<3>(sbase, w, ldsl, pk, p0, p1, o0[3], o1[3]);
    do_stage<4>(sbase, w, ldsl, pk, p0, p1, o0[4], o1[4]);
    do_stage<5>(sbase, w, ldsl, pk, p0, p1, o0[5], o1[5]);
    do_stage<6>(sbase, w, ldsl, pk, p0, p1, o0[6], o1[6]);
    do_stage<7>(sbase, w, ldsl, pk, p0, p1, o0[7], o1[7]);

    // out[n, c, h*8+a, w*8 + b], b=0..7 contiguous -> 2x b128 NT stores/channel.
    const size_t outC = (size_t)(UPF * H_) * (size_t)(UPF * W_);  // 512*1024
    float* r0 = out + (size_t)(n * 2) * outC
              + (size_t)(h * UPF + a) * (size_t)(UPF * W_) + (size_t)w * UPF;
    float* r1 = r0 + outC;
    const v4f x0 = {o0[0], o0[1], o0[2], o0[3]};
    const v4f x1 = {o0[4], o0[5], o0[6], o0[7]};
    const v4f y0 = {o1[0], o1[1], o1[2], o1[3]};
    const v4f y1 = {o1[4], o1[5], o1[6], o1[7]};
    __builtin_nontemporal_store(x0, (v4f*)r0);
    __builtin_nontemporal_store(x1, (v4f*)(r0 + 4));
    __builtin_nontemporal_store(y0, (v4f*)r1);
    __builtin_nontemporal_store(y1, (v4f*)(r1 + 4));
}

extern "C" void kernel_launch(void* const* d_in, const int* in_sizes, int n_in,
                              void* d_out, int out_size, void* d_ws, size_t ws_size,
                              hipStream_t stream) {
    const float* flow = (const float*)d_in[0];  // [N,2,64,128] fp32
    const float* mask = (const float*)d_in[1];  // [N,576,64,128] fp32
    float* out = (float*)d_out;                 // [N,2,512,1024] fp32

    const int N = in_sizes[0] / (2 * H_ * W_);  // = 8
    dim3 grid(N * H_ * UPF);   // 4096 blocks: one (n, h, a) sub-row each
    dim3 block(NT);            // 128 threads = 4 wave32s, thread = w
    FlowUpSampler_kernel<<<grid, block, 0, stream>>>(flow, mask, out);
}